// MPJPE_Loss_37374805410295
// MI455X (gfx1250) — compile-verified
//
#include <hip/hip_runtime.h>
#include <hip/hip_bf16.h>
#include <math.h>

// CDNA5 / gfx1250: wave32, WMMA 16x16x4 F32 used as an exact fp32 reduction engine.
typedef __attribute__((ext_vector_type(2))) float v2f;  // A / B matrix: 2 VGPRs per lane
typedef __attribute__((ext_vector_type(8))) float v8f;  // C / D matrix: 8 VGPRs per lane

// One wave32 computes l3D = (1/B) * sum |joint3d - pred| over n elements.
// WMMA D = A x 1s + C accumulates 64 |diff| values per instruction into a
// 16x16 f32 accumulator; each of the 16 columns holds identical row-sums,
// so total accumulator sum == 16 * sum(|diff|). Tail elements (n % 64) are
// pre-scaled by 16 so a single exact *(1/(16*B)) rescale finishes the job.
__global__ __launch_bounds__(32)
void l3d_wmma_reduce(const float* __restrict__ pred,
                     const float* __restrict__ joint3d,
                     float* __restrict__ out,
                     int n_tiles, int tail_start, int n, float scale) {
    const int lane  = threadIdx.x & 31;
    const int m     = lane & 15;          // A-matrix row held by this lane
    const int kbase = (lane >> 4) << 1;   // K=0..1 (lanes 0-15) or K=2..3 (lanes 16-31)
    const int base  = m * 4 + kbase;      // linear offset of this lane's float2 in a 64-elem tile

    v2f ones; ones.x = 1.0f; ones.y = 1.0f;   // B matrix: all ones (layout-invariant)
    v8f acc = {};                              // C/D accumulator, zero-init

    // EXEC is all-ones here (32 threads, no divergence) as WMMA requires.
    for (int t = 0; t < n_tiles; ++t) {
        const int idx = t * 64 + base;
        v2f p = *(const v2f*)(pred    + idx);
        v2f j = *(const v2f*)(joint3d + idx);
        v2f a;
        a.x = fabsf(j.x - p.x);
        a.y = fabsf(j.y - p.y);
        // D = A x B + C ; 8 args: (neg_a, A, neg_b, B, c_mod, C, reuse_a, reuse_b)
        acc = __builtin_amdgcn_wmma_f32_16x16x4_f32(
                  false, a, false, ones, (short)0, acc, false, false);
    }

    // Per-lane partial: sum of this lane's 8 accumulator rows (one column each half).
    float s = acc[0] + acc[1] + acc[2] + acc[3] + acc[4] + acc[5] + acc[6] + acc[7];

    // Tail (n % 64 elements): scale by 16 to match the 16x-replicated WMMA total.
    float t_acc = 0.0f;
    for (int i = tail_start + lane; i < n; i += 32)
        t_acc += fabsf(joint3d[i] - pred[i]);
    s += 16.0f * t_acc;

    // Wave32 butterfly reduction.
    for (int off = 16; off > 0; off >>= 1)
        s += __shfl_xor(s, off, 32);

    if (lane == 0)
        *out = s * scale;   // scale = 1 / (16 * B): exact power-of-two rescale
}

extern "C" void kernel_launch(void* const* d_in, const int* in_sizes, int n_in,
                              void* d_out, int out_size, void* d_ws, size_t ws_size,
                              hipStream_t stream) {
    (void)n_in; (void)out_size; (void)d_ws; (void)ws_size;
    // setup_inputs order: pred (B,17,3), joint3d (B,17,3), middle_out, heatmap.
    // loss = l3D + 0.0*Lint == l3D  ->  middle_out / heatmap are dead inputs.
    const float* pred    = (const float*)d_in[0];
    const float* joint3d = (const float*)d_in[1];
    float*       out     = (float*)d_out;

    const int n = in_sizes[0];                 // 64*17*3 = 3264
    const int n_tiles = n / 64;                // 51 full WMMA tiles
    const int tail_start = n_tiles * 64;       // 3264 -> empty tail
    const int B = n / (17 * 3);                // batch = 64
    const float scale = 1.0f / (16.0f * (float)B);  // 1/1024: exact in fp32

    l3d_wmma_reduce<<<1, 32, 0, stream>>>(pred, joint3d, out,
                                          n_tiles, tail_start, n, scale);
}